// MaskedDecoderLayer_86199993630777
// MI455X (gfx1250) — compile-verified
//
#include <hip/hip_runtime.h>
#include <math.h>

// ---------------------------------------------------------------------------
// MaskedDecoderLayer on gfx1250 (MI455X), fp32 via V_WMMA_F32_16X16X4_F32.
// B=16, NQ=100, NM=5440, NHR=4096, D=256, H=8, dh=32, FF=2048.
// GEMMs use 2x4 register blocking (32x64 output per wave): per k-step,
// 6 float2 loads feed 8 WMMAs (load:wmma = 0.75 vs 2.0 unblocked).
// ---------------------------------------------------------------------------

#define Bc   16
#define NQc  100
#define NMc  5440
#define NHRc 4096
#define Dc   256
#define Hc   8
#define DHc  32
#define FFc  2048

typedef __attribute__((ext_vector_type(2))) float v2f;
typedef __attribute__((ext_vector_type(8))) float v8f;

// D = A(16x4) * B(4x16) + C, fp32 in/out. Lane layout (wave32):
//  A: lane m = lane%16, VGPR j holds A[m][2*(lane/16)+j]
//  B: lane n = lane%16, VGPR j holds B[2*(lane/16)+j][n]
//  C/D: lane n = lane%16, VGPR r holds C[r + 8*(lane/16)][n]
static __device__ __forceinline__ v8f wmma_f32_k4(v2f a, v2f b, v8f c) {
  return __builtin_amdgcn_wmma_f32_16x16x4_f32(
      /*neg_a=*/false, a, /*neg_b=*/false, b,
      /*c_mod=*/(short)0, c, /*reuse_a=*/false, /*reuse_b=*/false);
}

// ---------------------------------------------------------------------------
// Fused LayerNorm (+ optional query_pos add). One 256-thread block per row.
// ---------------------------------------------------------------------------
__global__ __launch_bounds__(256)
void ln_addpos_kernel(const float* __restrict__ x, const float* __restrict__ w,
                      const float* __restrict__ bb, const float* __restrict__ pos,
                      float* __restrict__ out_ln, float* __restrict__ out_pos) {
  int row = blockIdx.x;
  int t = threadIdx.x;
  __shared__ float red[256];
  size_t idx = (size_t)row * Dc + t;
  float xv = x[idx];
  red[t] = xv;
  __syncthreads();
  for (int s = 128; s > 0; s >>= 1) {
    if (t < s) red[t] += red[t + s];
    __syncthreads();
  }
  float mean = red[0] * (1.0f / Dc);
  __syncthreads();
  float d = xv - mean;
  red[t] = d * d;
  __syncthreads();
  for (int s = 128; s > 0; s >>= 1) {
    if (t < s) red[t] += red[t + s];
    __syncthreads();
  }
  float var = red[0] * (1.0f / Dc);
  float y = d * rsqrtf(var + 1e-5f) * w[t] + bb[t];
  if (out_ln) out_ln[idx] = y;
  if (out_pos) out_pos[idx] = y + pos[idx];
}

// ---------------------------------------------------------------------------
// Y[R,N] = act(X[R,K] @ W[N,K]^T + bias) (+ residual).
// One wave computes a 32x64 block (2 row-tiles x 4 col-tiles).
// R multiple of 32, N multiple of 64, K multiple of 4. act: 0=none, 1=GELU.
// ---------------------------------------------------------------------------
__global__ __launch_bounds__(32)
void gemm_xwT_wmma(const float* __restrict__ X, const float* __restrict__ W,
                   const float* __restrict__ bias, const float* __restrict__ res,
                   float* __restrict__ Y, int K, int N, int act) {
  int lane = threadIdx.x;
  int lo = lane & 15;
  int kk = (lane >> 4) * 2;
  size_t r0 = (size_t)blockIdx.x * 32;
  size_t n0 = (size_t)blockIdx.y * 64;
  const float* xr0 = X + (r0 + lo) * (size_t)K;
  const float* xr1 = X + (r0 + 16 + lo) * (size_t)K;
  const float* wr0 = W + (n0 + lo) * (size_t)K;
  const float* wr1 = wr0 + 16 * (size_t)K;
  const float* wr2 = wr0 + 32 * (size_t)K;
  const float* wr3 = wr0 + 48 * (size_t)K;

  v8f c00 = {}, c01 = {}, c02 = {}, c03 = {};
  v8f c10 = {}, c11 = {}, c12 = {}, c13 = {};
#pragma unroll 2
  for (int kb = 0; kb < K; kb += 4) {
    v2f a0 = *(const v2f*)(xr0 + kb + kk);
    v2f a1 = *(const v2f*)(xr1 + kb + kk);
    v2f b0 = *(const v2f*)(wr0 + kb + kk);
    v2f b1 = *(const v2f*)(wr1 + kb + kk);
    v2f b2 = *(const v2f*)(wr2 + kb + kk);
    v2f b3 = *(const v2f*)(wr3 + kb + kk);
    c00 = wmma_f32_k4(a0, b0, c00);
    c01 = wmma_f32_k4(a0, b1, c01);
    c02 = wmma_f32_k4(a0, b2, c02);
    c03 = wmma_f32_k4(a0, b3, c03);
    c10 = wmma_f32_k4(a1, b0, c10);
    c11 = wmma_f32_k4(a1, b1, c11);
    c12 = wmma_f32_k4(a1, b2, c12);
    c13 = wmma_f32_k4(a1, b3, c13);
  }

  float bn0 = bias ? bias[n0 + lo] : 0.0f;
  float bn1 = bias ? bias[n0 + 16 + lo] : 0.0f;
  float bn2 = bias ? bias[n0 + 32 + lo] : 0.0f;
  float bn3 = bias ? bias[n0 + 48 + lo] : 0.0f;
  int hi8 = (lane >> 4) * 8;

  v8f* acc[2][4] = {{&c00, &c01, &c02, &c03}, {&c10, &c11, &c12, &c13}};
  float bn[4] = {bn0, bn1, bn2, bn3};
#pragma unroll
  for (int i = 0; i < 2; ++i) {
#pragma unroll
    for (int r = 0; r < 8; ++r) {
      size_t row = r0 + (size_t)(16 * i + r + hi8);
      size_t rowBase = row * (size_t)N + n0 + lo;
#pragma unroll
      for (int j = 0; j < 4; ++j) {
        float v = (*acc[i][j])[r] + bn[j];
        if (act == 1) v = 0.5f * v * (1.0f + erff(v * 0.70710678118654752f));
        size_t idx = rowBase + 16 * j;
        if (res) v += res[idx];
        Y[idx] = v;
      }
    }
  }
}

// ---------------------------------------------------------------------------
// Flash attention, fp32 WMMA. One wave per (q-tile of 16, head, batch).
// Q/K/V are [B, rows, D] with per-head slice of width dh=32. NKV keys per
// batch (buffer rows per batch == NKV). Optional additive bias [B,NQ,NKV]
// (broadcast over heads). Online softmax with LDS 16x16 score tile.
// ---------------------------------------------------------------------------
__global__ __launch_bounds__(32)
void flash_attn_wmma(const float* __restrict__ Q, const float* __restrict__ Kb,
                     const float* __restrict__ Vb, const float* __restrict__ bias,
                     float* __restrict__ O, int NKV) {
  int lane = threadIdx.x;
  int lo = lane & 15;
  int hi = lane >> 4;
  int kk = hi * 2;
  int qt = blockIdx.x, h = blockIdx.y, b = blockIdx.z;
  int q0 = qt * 16;
  __shared__ float sS[16][17];
  __shared__ float sFac[16];
  __shared__ float sL[16];

  int qA = q0 + lo;
  int qAc = qA < NQc ? qA : NQc - 1;                 // clamp padded q rows
  const float* qrow = Q + ((size_t)b * NQc + qAc) * Dc + h * DHc;
  const float scale = 0.17677669529663687f;          // 1/sqrt(32)

  float m_run = -3.0e38f, l_run = 0.0f;              // lanes 0..15 own row=lane
  v8f o0 = {};
  v8f o1 = {};
  int ntiles = (NKV + 15) >> 4;

  for (int mt = 0; mt < ntiles; ++mt) {
    int m0 = mt * 16;
    int mB = m0 + lo;
    int mBc = mB < NKV ? mB : NKV - 1;               // clamp padded key rows
    const float* krow = Kb + ((size_t)b * NKV + mBc) * Dc + h * DHc;

    // S tile = Q(16x32) * K^T(32x16)
    v8f cs = {};
#pragma unroll
    for (int kb = 0; kb < DHc; kb += 4) {
      v2f a = *(const v2f*)(qrow + kb + kk);
      v2f bf = *(const v2f*)(krow + kb + kk);
      cs = wmma_f32_k4(a, bf, cs);
    }
    // scale + prior bias, stage to LDS
#pragma unroll
    for (int r = 0; r < 8; ++r) {
      int qr = q0 + r + hi * 8;
      int qrc = qr < NQc ? qr : NQc - 1;
      float sv = cs[r] * scale;
      if (bias) sv += bias[((size_t)b * NQc + qrc) * (size_t)NKV + mBc];
      sS[r + hi * 8][lo] = sv;
    }
    __syncthreads();

    // per-row online softmax stats; write P back into sS
    if (lane < 16) {
      int i = lane;
      float tmax = -3.0e38f;
      float sv[16];
#pragma unroll
      for (int c2 = 0; c2 < 16; ++c2) {
        float s = (m0 + c2 < NKV) ? sS[i][c2] : -3.0e38f;
        sv[c2] = s;
        tmax = fmaxf(tmax, s);
      }
      float mnew = fmaxf(m_run, tmax);
      float fac = expf(m_run - mnew);
      float ps = 0.0f;
#pragma unroll
      for (int c2 = 0; c2 < 16; ++c2) {
        float p = (m0 + c2 < NKV) ? expf(sv[c2] - mnew) : 0.0f;
        sS[i][c2] = p;
        ps += p;
      }
      l_run = l_run * fac + ps;
      m_run = mnew;
      sFac[i] = fac;
    }
    __syncthreads();

    // rescale accumulators by per-row factor
#pragma unroll
    for (int r = 0; r < 8; ++r) {
      float f = sFac[r + hi * 8];
      o0[r] *= f;
      o1[r] *= f;
    }
    // O += P(16x16) * V(16x32)  (two 16-wide d halves)
#pragma unroll
    for (int kb = 0; kb < 16; kb += 4) {
      v2f a;
      a.x = sS[lo][kb + kk];
      a.y = sS[lo][kb + kk + 1];
      int k0 = m0 + kb + kk;
      int k0c = k0 < NKV ? k0 : NKV - 1;
      int k1c = (k0 + 1) < NKV ? (k0 + 1) : NKV - 1;
      const float* v0 = Vb + ((size_t)b * NKV + k0c) * Dc + h * DHc;
      const float* v1 = Vb + ((size_t)b * NKV + k1c) * Dc + h * DHc;
      v2f b0; b0.x = v0[lo];      b0.y = v1[lo];
      v2f b1; b1.x = v0[16 + lo]; b1.y = v1[16 + lo];
      o0 = wmma_f32_k4(a, b0, o0);
      o1 = wmma_f32_k4(a, b1, o1);
    }
    __syncthreads();
  }

  if (lane < 16) sL[lane] = l_run;
  __syncthreads();
#pragma unroll
  for (int r = 0; r < 8; ++r) {
    int qr = q0 + r + hi * 8;
    if (qr < NQc) {
      float invl = 1.0f / sL[r + hi * 8];
      size_t base = ((size_t)b * NQc + qr) * Dc + h * DHc;
      O[base + lo] = o0[r] * invl;
      O[base + 16 + lo] = o1[r] * invl;
    }
  }
}

// ---------------------------------------------------------------------------
// mask_logits[b,q,n] = T[b,q,:] . HR[b,n,:]   (batched, q padded 100->112)
// One wave computes 16 x 64 (1 q-tile x 4 col-tiles), reusing the A fragment.
// ---------------------------------------------------------------------------
__global__ __launch_bounds__(32)
void mask_logits_wmma(const float* __restrict__ T, const float* __restrict__ HR,
                      float* __restrict__ out) {
  int lane = threadIdx.x;
  int lo = lane & 15;
  int hi = lane >> 4;
  int kk = hi * 2;
  int qt = blockIdx.x;
  size_t n0 = (size_t)blockIdx.y * 64;
  int b = blockIdx.z;
  int q = qt * 16 + lo;
  int qc = q < NQc ? q : NQc - 1;
  const float* tr = T + ((size_t)b * NQc + qc) * Dc;
  const float* hr0 = HR + ((size_t)b * NHRc + n0 + lo) * Dc;
  const float* hr1 = hr0 + 16 * (size_t)Dc;
  const float* hr2 = hr0 + 32 * (size_t)Dc;
  const float* hr3 = hr0 + 48 * (size_t)Dc;
  v8f c0 = {}, c1 = {}, c2 = {}, c3 = {};
#pragma unroll 2
  for (int kb = 0; kb < Dc; kb += 4) {
    v2f a = *(const v2f*)(tr + kb + kk);
    v2f b0 = *(const v2f*)(hr0 + kb + kk);
    v2f b1 = *(const v2f*)(hr1 + kb + kk);
    v2f b2 = *(const v2f*)(hr2 + kb + kk);
    v2f b3 = *(const v2f*)(hr3 + kb + kk);
    c0 = wmma_f32_k4(a, b0, c0);
    c1 = wmma_f32_k4(a, b1, c1);
    c2 = wmma_f32_k4(a, b2, c2);
    c3 = wmma_f32_k4(a, b3, c3);
  }
  int hi8 = hi * 8;
#pragma unroll
  for (int r = 0; r < 8; ++r) {
    int qr = qt * 16 + r + hi8;
    if (qr < NQc) {
      size_t base = ((size_t)b * NQc + qr) * (size_t)NHRc + n0 + lo;
      out[base] = c0[r];
      out[base + 16] = c1[r];
      out[base + 32] = c2[r];
      out[base + 48] = c3[r];
    }
  }
}

// ---------------------------------------------------------------------------
// Host-side launch sequence
// ---------------------------------------------------------------------------
extern "C" void kernel_launch(void* const* d_in, const int* in_sizes, int n_in,
                              void* d_out, int out_size, void* d_ws, size_t ws_size,
                              hipStream_t stream) {
  (void)in_sizes; (void)n_in; (void)out_size; (void)ws_size;

  const float* tgt      = (const float*)d_in[0];
  const float* memory   = (const float*)d_in[1];
  const float* memhr    = (const float*)d_in[2];
  const float* pbias    = (const float*)d_in[3];
  const float* qpos     = (const float*)d_in[4];
  const float* lncw     = (const float*)d_in[5];
  const float* lncb     = (const float*)d_in[6];
  const float* q_w      = (const float*)d_in[7];
  const float* q_b      = (const float*)d_in[8];
  const float* k_w      = (const float*)d_in[9];
  const float* k_b      = (const float*)d_in[10];
  const float* v_w      = (const float*)d_in[11];
  const float* v_b      = (const float*)d_in[12];
  const float* out_w    = (const float*)d_in[13];
  const float* out_b    = (const float*)d_in[14];
  const float* lnsw     = (const float*)d_in[15];
  const float* lnsb     = (const float*)d_in[16];
  const float* sa_in_w  = (const float*)d_in[17];
  const float* sa_in_b  = (const float*)d_in[18];
  const float* sa_out_w = (const float*)d_in[19];
  const float* sa_out_b = (const float*)d_in[20];
  const float* lnfw     = (const float*)d_in[21];
  const float* lnfb     = (const float*)d_in[22];
  const float* l1w      = (const float*)d_in[23];
  const float* l1b      = (const float*)d_in[24];
  const float* l2w      = (const float*)d_in[25];
  const float* l2b      = (const float*)d_in[26];
  const float* me_w     = (const float*)d_in[27];
  const float* me_b     = (const float*)d_in[28];

  const size_t S_TOK = (size_t)Bc * NQc * Dc;   // 409,600
  const size_t S_KV  = (size_t)Bc * NMc * Dc;   // 22,282,240
  const size_t S_FF  = (size_t)Bc * NQc * FFc;  // 3,276,800

  float* outT    = (float*)d_out;               // final tgt [B,NQ,D]
  float* outMask = outT + S_TOK;                // mask_logits [B,NQ,NHR]

  float* ws   = (float*)d_ws;
  float* xq   = ws;              // ln_cross(tgt) + query_pos
  float* qbuf = xq + S_TOK;
  float* kbuf = qbuf + S_TOK;
  float* vbuf = kbuf + S_KV;
  float* obuf = vbuf + S_KV;
  float* t1   = obuf + S_TOK;
  float* z    = t1 + S_TOK;
  float* zq   = z + S_TOK;
  float* qs   = zq + S_TOK;
  float* ks   = qs + S_TOK;
  float* vs   = ks + S_TOK;
  float* so   = vs + S_TOK;
  float* t2   = so + S_TOK;
  float* yl   = t2 + S_TOK;
  float* ff   = yl + S_TOK;
  float* tm   = ff + S_FF;

  dim3 blk32(32), blk256(256);
  const int rowsTok = Bc * NQc;   // 1600  (divisible by 32)
  const int rowsKV  = Bc * NMc;   // 87040 (divisible by 32)

  // ---- cross-attention ----
  ln_addpos_kernel<<<rowsTok, blk256, 0, stream>>>(tgt, lncw, lncb, qpos, nullptr, xq);
  gemm_xwT_wmma<<<dim3(rowsTok / 32, Dc / 64), blk32, 0, stream>>>(xq, q_w, q_b, nullptr, qbuf, Dc, Dc, 0);
  gemm_xwT_wmma<<<dim3(rowsKV / 32, Dc / 64), blk32, 0, stream>>>(memory, k_w, k_b, nullptr, kbuf, Dc, Dc, 0);
  gemm_xwT_wmma<<<dim3(rowsKV / 32, Dc / 64), blk32, 0, stream>>>(memory, v_w, v_b, nullptr, vbuf, Dc, Dc, 0);
  flash_attn_wmma<<<dim3((NQc + 15) / 16, Hc, Bc), blk32, 0, stream>>>(qbuf, kbuf, vbuf, pbias, obuf, NMc);
  gemm_xwT_wmma<<<dim3(rowsTok / 32, Dc / 64), blk32, 0, stream>>>(obuf, out_w, out_b, tgt, t1, Dc, Dc, 0);

  // ---- self-attention ----
  ln_addpos_kernel<<<rowsTok, blk256, 0, stream>>>(t1, lnsw, lnsb, qpos, z, zq);
  gemm_xwT_wmma<<<dim3(rowsTok / 32, Dc / 64), blk32, 0, stream>>>(zq, sa_in_w, sa_in_b, nullptr, qs, Dc, Dc, 0);
  gemm_xwT_wmma<<<dim3(rowsTok / 32, Dc / 64), blk32, 0, stream>>>(zq, sa_in_w + (size_t)Dc * Dc, sa_in_b + Dc, nullptr, ks, Dc, Dc, 0);
  gemm_xwT_wmma<<<dim3(rowsTok / 32, Dc / 64), blk32, 0, stream>>>(z, sa_in_w + 2 * (size_t)Dc * Dc, sa_in_b + 2 * Dc, nullptr, vs, Dc, Dc, 0);
  flash_attn_wmma<<<dim3((NQc + 15) / 16, Hc, Bc), blk32, 0, stream>>>(qs, ks, vs, nullptr, so, NQc);
  gemm_xwT_wmma<<<dim3(rowsTok / 32, Dc / 64), blk32, 0, stream>>>(so, sa_out_w, sa_out_b, t1, t2, Dc, Dc, 0);

  // ---- FFN ----
  ln_addpos_kernel<<<rowsTok, blk256, 0, stream>>>(t2, lnfw, lnfb, nullptr, yl, nullptr);
  gemm_xwT_wmma<<<dim3(rowsTok / 32, FFc / 64), blk32, 0, stream>>>(yl, l1w, l1b, nullptr, ff, Dc, FFc, 1);
  gemm_xwT_wmma<<<dim3(rowsTok / 32, Dc / 64), blk32, 0, stream>>>(ff, l2w, l2b, t2, outT, FFc, Dc, 0);

  // ---- mask branch ----
  gemm_xwT_wmma<<<dim3(rowsTok / 32, Dc / 64), blk32, 0, stream>>>(outT, me_w, me_b, nullptr, tm, Dc, Dc, 0);
  mask_logits_wmma<<<dim3((NQc + 15) / 16, NHRc / 64, Bc), blk32, 0, stream>>>(tm, memhr, outMask);
}